// HausdorffDTLoss_66340064854729
// MI455X (gfx1250) — compile-verified
//
#include <hip/hip_runtime.h>
#include <hip/hip_bf16.h>

// ---------------- problem constants ----------------
#define BATCH 8
#define HH 384
#define WW 384
#define NPIX ((size_t)BATCH * HH * WW)   // 1,179,648 per plane
#define BIGV 1000000.0f                  // matches reference sentinel

typedef __attribute__((ext_vector_type(2))) float v2f;
typedef __attribute__((ext_vector_type(8))) float v8f;

// address-space-qualified pointee types for the async-to-LDS builtin
typedef __attribute__((address_space(1))) int as1_int_t;   // global
typedef __attribute__((address_space(3))) int as3_int_t;   // LDS

// ---------------- CDNA5 feature probes ----------------
#if defined(__has_builtin)
#if __has_builtin(__builtin_amdgcn_global_load_async_to_lds_b32) && \
    __has_builtin(__builtin_amdgcn_s_wait_asynccnt)
#define USE_ASYNC_LDS 1
#endif
#if __has_builtin(__builtin_amdgcn_wmma_f32_16x16x4_f32)
#define USE_WMMA_REDUCE 1
#endif
#endif

// Sum of a float over all 32 lanes of a wave.
// WMMA trick: A(16x4) holds the 32 partials (lane L -> A[L%16][L<16?0:2]),
// B = all-ones => D[m][n] = p_m + p_{m+16}.  Lanes 0-15 hold row-sums m=0..7
// in D VGPRs 0..7, lanes 16-31 hold m=8..15; add the two halves with one xor-shuffle.
__device__ __forceinline__ float wave_sum(float p) {
#if defined(USE_WMMA_REDUCE)
  v2f a; a[0] = p; a[1] = 0.0f;
  v2f bo; bo[0] = 1.0f; bo[1] = 1.0f;
  v8f c = {0.f, 0.f, 0.f, 0.f, 0.f, 0.f, 0.f, 0.f};
  v8f d = __builtin_amdgcn_wmma_f32_16x16x4_f32(
      /*neg_a=*/false, a, /*neg_b=*/false, bo,
      /*c_mod=*/(short)0, c, /*reuse_a=*/false, /*reuse_b=*/false);
  float t = ((d[0] + d[1]) + (d[2] + d[3])) + ((d[4] + d[5]) + (d[6] + d[7]));
  return t + __shfl_xor(t, 16, 32);
#else
  for (int off = 16; off > 0; off >>= 1) p += __shfl_xor(p, off, 32);
  return p;
#endif
}

// ---------------- kernel 0: zero partials + flags ----------------
__global__ void hdt_init(float* __restrict__ partials, int* __restrict__ flags) {
  int idx = blockIdx.x * blockDim.x + threadIdx.x;
  if (idx < BATCH * HH) partials[idx] = 0.0f;
  if (idx < 16) flags[idx] = 0;
}

// ---------------- kernel 1: per-column 1D distance, squared ----------------
// For one image plane `img` (pred or target): g2f = squared vertical distance
// to nearest BACKGROUND pixel (seeds = !m, used by edt(fg));
// g2b = squared vertical distance to nearest FOREGROUND pixel (edt(bg)).
// One thread per column; two passes (down stores fwd, up folds in bwd and squares).
__global__ void hdt_coldist(const float* __restrict__ img,
                            float* __restrict__ g2f, float* __restrict__ g2b,
                            int* __restrict__ flags, int which) {
  const int b = blockIdx.x / (WW / 128);
  const int j = (blockIdx.x % (WW / 128)) * 128 + threadIdx.x;
  const size_t base = (size_t)b * HH * WW + j;
  const float* col = img + base;
  float* gf = g2f + base;
  float* gb = g2b + base;

  unsigned bits[HH / 32];
#pragma unroll
  for (int w = 0; w < HH / 32; ++w) bits[w] = 0u;

  float cf = BIGV, cb = BIGV;
  unsigned any = 0u;
  for (int i = 0; i < HH; ++i) {
    float v = col[(size_t)i * WW];
    unsigned m = (v >= 0.5f) ? 1u : 0u;
    bits[i >> 5] |= m << (i & 31);
    any |= m;
    float xf = m ? BIGV : 0.0f;   // seed (bg) -> 0
    float xb = m ? 0.0f : BIGV;   // seed (fg) -> 0
    cf = fminf(xf, cf + 1.0f);    // identical recurrence to the jax scan
    cb = fminf(xb, cb + 1.0f);
    gf[(size_t)i * WW] = cf;      // forward pass result
    gb[(size_t)i * WW] = cb;
  }
  cf = BIGV; cb = BIGV;
  for (int i = HH - 1; i >= 0; --i) {
    unsigned m = (bits[i >> 5] >> (i & 31)) & 1u;
    float xf = m ? BIGV : 0.0f;
    float xb = m ? 0.0f : BIGV;
    cf = fminf(xf, cf + 1.0f);
    cb = fminf(xb, cb + 1.0f);
    float g0 = fminf(gf[(size_t)i * WW], cf);
    float g1 = fminf(gb[(size_t)i * WW], cb);
    gf[(size_t)i * WW] = g0 * g0; // store squared distances
    gb[(size_t)i * WW] = g1 * g1;
  }
  if (any) atomicOr(&flags[which * BATCH + b], 1);  // fg.any() per image
}

// ---------------- kernel 2: row envelope + masked loss partials ----------------
// One block per (image, row).  Stage g2 rows into LDS (async-to-LDS when the
// toolchain exposes it), brute-force exact lower envelope over source columns,
// accumulate pe * (d2_fg + d2_bg) gated by the per-image fg.any() flag.
__global__ void hdt_envelope(const float* __restrict__ pred,
                             const float* __restrict__ target,
                             const float* __restrict__ g2f,
                             const float* __restrict__ g2b,
                             const int* __restrict__ flags,
                             float* __restrict__ partials, int which) {
  __shared__ float sgf[WW];
  __shared__ float sgb[WW];
  __shared__ float spe[WW];
  __shared__ float swave[4];

  const int b = blockIdx.x / HH;
  const int i = blockIdx.x % HH;
  const size_t row = ((size_t)b * HH + i) * WW;
  const int tid = threadIdx.x;

#pragma unroll
  for (int k = 0; k < 3; ++k) {
    int t = tid + k * 128;
    float pv = pred[row + t];
    float tv = target[row + t];
    spe[t] = ((pv >= 0.5f) != (tv >= 0.5f)) ? 1.0f : 0.0f;  // (pb - tb)^2
  }

#if defined(USE_ASYNC_LDS)
#pragma unroll
  for (int k = 0; k < 3; ++k) {
    int t = tid + k * 128;
    __builtin_amdgcn_global_load_async_to_lds_b32(
        (as1_int_t*)(g2f + row + t), (as3_int_t*)(&sgf[t]), 0, 0);
    __builtin_amdgcn_global_load_async_to_lds_b32(
        (as1_int_t*)(g2b + row + t), (as3_int_t*)(&sgb[t]), 0, 0);
  }
  __builtin_amdgcn_s_wait_asynccnt(0);
#else
#pragma unroll
  for (int k = 0; k < 3; ++k) {
    int t = tid + k * 128;
    sgf[t] = g2f[row + t];
    sgb[t] = g2b[row + t];
  }
#endif
  __syncthreads();

  const float flag = flags[which * BATCH + b] ? 1.0f : 0.0f;

  float acc = 0.0f;
#pragma unroll
  for (int k = 0; k < 3; ++k) {
    const int jd = tid + k * 128;
    float mf = 3.0e38f, mb = 3.0e38f;
#pragma unroll 4
    for (int js = 0; js < WW; ++js) {
      float diff = (float)(jd - js);
      float o2 = diff * diff;               // exact (< 2^24)
      mf = fminf(mf, sgf[js] + o2);
      mb = fminf(mb, sgb[js] + o2);
    }
    acc += spe[jd] * (mf + mb);             // d2_fg + d2_bg (disjoint support)
  }
  acc *= flag;

  // EXEC is all-ones here (no divergence above) -> WMMA reduction is legal.
  float ws = wave_sum(acc);
  const int wave = tid >> 5, lane = tid & 31;
  if (lane == 0) swave[wave] = ws;
  __syncthreads();
  if (tid == 0)
    partials[blockIdx.x] += (swave[0] + swave[1]) + (swave[2] + swave[3]);
}

// ---------------- kernel 3: deterministic final reduction ----------------
__global__ void hdt_final(const float* __restrict__ partials, float* __restrict__ out) {
  __shared__ float red[256];
  const int tid = threadIdx.x;
  float s = 0.0f;
  for (int idx = tid; idx < BATCH * HH; idx += 256) s += partials[idx];
  red[tid] = s;
  __syncthreads();
  for (int off = 128; off > 0; off >>= 1) {
    if (tid < off) red[tid] += red[tid + off];
    __syncthreads();
  }
  if (tid == 0) out[0] = red[0] * (1.0f / (float)NPIX);   // mean over B*C*H*W
}

// ---------------- launcher ----------------
extern "C" void kernel_launch(void* const* d_in, const int* in_sizes, int n_in,
                              void* d_out, int out_size, void* d_ws, size_t ws_size,
                              hipStream_t stream) {
  (void)in_sizes; (void)n_in; (void)out_size; (void)ws_size;
  const float* pred = (const float*)d_in[0];
  const float* target = (const float*)d_in[1];
  float* ws = (float*)d_ws;

  // workspace layout (floats): [g2f NPIX][g2b NPIX][partials B*H][flags 16 ints]
  float* g2f = ws;
  float* g2b = ws + NPIX;
  float* partials = ws + 2 * NPIX;
  int* flags = (int*)(ws + 2 * NPIX + BATCH * HH);

  hdt_init<<<(BATCH * HH + 255) / 256, 256, 0, stream>>>(partials, flags);

  const int colBlocks = BATCH * (WW / 128);   // 24 blocks x 128 threads
  const int rowBlocks = BATCH * HH;           // 3072 blocks x 128 threads

  // phase 0: pred distance field, masked by pred_error
  hdt_coldist<<<colBlocks, 128, 0, stream>>>(pred, g2f, g2b, flags, 0);
  hdt_envelope<<<rowBlocks, 128, 0, stream>>>(pred, target, g2f, g2b, flags,
                                              partials, 0);
  // phase 1: target distance field (reuses the same g2 planes)
  hdt_coldist<<<colBlocks, 128, 0, stream>>>(target, g2f, g2b, flags, 1);
  hdt_envelope<<<rowBlocks, 128, 0, stream>>>(pred, target, g2f, g2b, flags,
                                              partials, 1);

  hdt_final<<<1, 256, 0, stream>>>(partials, (float*)d_out);
}